// ToRGBLayer_11355893531018
// MI455X (gfx1250) — compile-verified
//
#include <hip/hip_runtime.h>
#include <hip/hip_bf16.h>

typedef __attribute__((ext_vector_type(2))) float v2f;
typedef __attribute__((ext_vector_type(8))) float v8f;

#define BB 4
#define FF 98304
#define NN (BB * FF)          // 393216 real faces
#define C_IN 256
#define C_OUT 3
#define W_DIM 512
#define K2 4
#define KTOT (C_IN * K2)      // 1024 = reduction length per face
#define NCHUNK (KTOT / 4)     // 256 WMMA k-steps of 4
#define TILES_PER_B (FF / 16) // 6144
#define GRID_X 192

// ws layout (floats):
//   [0 .. BB*NCHUNK*64)              : B-fragments, ((b*NCHUNK+chunk)*32+lane)*2+v
//   [BB*NCHUNK*64 .. +C_IN)          : one zero row for pad gathers
#define WS_BFRAG_FLOATS (BB * NCHUNK * 64) // 65536
#define WS_ZROW_OFF WS_BFRAG_FLOATS

// ---------------------------------------------------------------------------
// Prep: styles + modulated weights laid out as WMMA B-fragments, + zero row.
// One block per batch element b, 256 threads.
// ---------------------------------------------------------------------------
__global__ __launch_bounds__(256) void faceconv_prep_kernel(
    const float* __restrict__ w,       // [B, W_DIM]
    const float* __restrict__ aw,      // [C_IN, W_DIM]
    const float* __restrict__ ab,      // [C_IN]
    const float* __restrict__ weight,  // [C_OUT, C_IN, 1, K2]
    float* __restrict__ ws) {
  const int b = blockIdx.x;
  const int c = threadIdx.x; // 0..255
  __shared__ float styles[C_IN];

  const float inv_sqrt_wdim = 0.04419417382415922f; // 1/sqrt(512)
  const float weight_gain   = 0.03125f;             // 1/sqrt(256*4)

  float acc = 0.f;
  const float* wrow = w + (size_t)b * W_DIM;
  const float* arow = aw + (size_t)c * W_DIM;
  for (int d = 0; d < W_DIM; ++d) acc += wrow[d] * arow[d];
  styles[c] = (acc * inv_sqrt_wdim + ab[c]) * weight_gain;
  __syncthreads();

  // Emit B-fragment values. Assumed 32-bit 4x16 B layout (mirrors documented A
  // layout): VGPR0 lanes 0-15 -> K=j0, lanes 16-31 -> K=j0+2; VGPR1 -> j0+1/j0+3.
  for (int idx = threadIdx.x; idx < NCHUNK * 64; idx += blockDim.x) {
    int v = idx & 1;
    int lane = (idx >> 1) & 31;
    int chunk = idx >> 6;
    int j = chunk * 4 + ((lane >= 16) ? 2 : 0) + v; // flattened (k2, c) index
    int o = lane & 15;                              // output channel column
    int cc = j & (C_IN - 1);
    int k2 = j >> 8;
    float val = 0.f;
    if (o < C_OUT) val = weight[(o * C_IN + cc) * K2 + k2] * styles[cc];
    ws[((size_t)(b * NCHUNK + chunk) * 32 + lane) * 2 + v] = val;
  }
  // Zero row used as gather target for pad indices (ws is poisoned by harness).
  if (b == 0)
    for (int i = threadIdx.x; i < C_IN; i += blockDim.x)
      ws[WS_ZROW_OFF + i] = 0.f;
}

// ---------------------------------------------------------------------------
// Main: gathered face conv as D(16x16) = A(16x1024) * B(1024x16) via
// 256 x v_wmma_f32_16x16x4_f32 per 16-face tile, four accumulator chains
// (one per neighbor) for WMMA/memory ILP.
// Block = 128 threads (4 waves); blockIdx.y = batch b; B-frags staged in LDS.
// ---------------------------------------------------------------------------
__global__ __launch_bounds__(128) void faceconv_wmma_kernel(
    const float* __restrict__ x,     // [N, C_IN]
    const int* __restrict__ nbh,     // [N, K2]
    const float* __restrict__ ws,    // prep output
    const float* __restrict__ bias,  // [C_OUT]
    float* __restrict__ out) {       // [N, C_OUT]
  const int b = blockIdx.y;

  __shared__ float bfr[NCHUNK * 64]; // 64 KB: this b's B-fragments
  {
    const float* src = ws + (size_t)b * (NCHUNK * 64);
    for (int i = threadIdx.x; i < NCHUNK * 64; i += blockDim.x) bfr[i] = src[i];
  }
  __syncthreads();

  const int lane = threadIdx.x & 31;
  const int wave = threadIdx.x >> 5;
  const int half = lane >> 4;   // 0: K lanes {j0,j0+1}; 1: {j0+2,j0+3}
  const int m = lane & 15;      // face row within tile / output column
  const float* zrow = ws + WS_ZROW_OFF;

  const bool writer = (m < C_OUT);
  const float bv = writer ? bias[m] : 0.f;

  // Loop-invariant LDS fragment bases, one per neighbor k2.
  const float* bB0 = &bfr[(0 * 64 * 32 + lane) * 2];
  const float* bB1 = &bfr[(1 * 64 * 32 + lane) * 2];
  const float* bB2 = &bfr[(2 * 64 * 32 + lane) * 2];
  const float* bB3 = &bfr[(3 * 64 * 32 + lane) * 2];

  const int stride = gridDim.x * 4;
  for (int t = blockIdx.x * 4 + wave; t < TILES_PER_B; t += stride) {
    const int tile = b * TILES_PER_B + t;
    const int face = tile * 16 + m;

    // Per-lane gathered-row pointers for the 4 neighbors (pads -> zero row).
    const float* aptr[K2];
#pragma unroll
    for (int k2 = 0; k2 < K2; ++k2) {
      int idx = nbh[face * K2 + k2];
      const float* base = (idx < NN) ? (x + (size_t)idx * C_IN) : zrow;
      aptr[k2] = base + half * 2;
    }
    const float* p0 = aptr[0];
    const float* p1 = aptr[1];
    const float* p2 = aptr[2];
    const float* p3 = aptr[3];

    v8f acc0 = {};
    v8f acc1 = {};
    v8f acc2 = {};
    v8f acc3 = {};
#pragma unroll 4
    for (int cc = 0; cc < 64; ++cc) {
      v2f a0 = *(const v2f*)(p0 + cc * 4);
      v2f a1 = *(const v2f*)(p1 + cc * 4);
      v2f a2 = *(const v2f*)(p2 + cc * 4);
      v2f a3 = *(const v2f*)(p3 + cc * 4);
      v2f f0 = *(const v2f*)(bB0 + cc * 64);
      v2f f1 = *(const v2f*)(bB1 + cc * 64);
      v2f f2 = *(const v2f*)(bB2 + cc * 64);
      v2f f3 = *(const v2f*)(bB3 + cc * 64);
      acc0 = __builtin_amdgcn_wmma_f32_16x16x4_f32(
          false, a0, false, f0, (short)0, acc0, false, false);
      acc1 = __builtin_amdgcn_wmma_f32_16x16x4_f32(
          false, a1, false, f1, (short)0, acc1, false, false);
      acc2 = __builtin_amdgcn_wmma_f32_16x16x4_f32(
          false, a2, false, f2, (short)0, acc2, false, false);
      acc3 = __builtin_amdgcn_wmma_f32_16x16x4_f32(
          false, a3, false, f3, (short)0, acc3, false, false);
    }

    // D layout: VGPR v holds M = v + half*8, N = lane&15. Only N<3 are real.
    if (writer) {
#pragma unroll
      for (int v = 0; v < 8; ++v) {
        int M = v + half * 8;
        float val = (acc0[v] + acc1[v]) + (acc2[v] + acc3[v]) + bv;
        val = fminf(fmaxf(val, -256.f), 256.f);
        out[(size_t)(tile * 16 + M) * C_OUT + m] = val;
      }
    }
  }
}

// ---------------------------------------------------------------------------
extern "C" void kernel_launch(void* const* d_in, const int* in_sizes, int n_in,
                              void* d_out, int out_size, void* d_ws, size_t ws_size,
                              hipStream_t stream) {
  const float* x   = (const float*)d_in[0];  // [N, C_IN]
  const int* nbh   = (const int*)d_in[1];    // [N, K2]
  // d_in[2] face_is_pad, d_in[3] pad_size: structurally known (pads are idx>=N)
  const float* w   = (const float*)d_in[4];  // [B, W_DIM]
  const float* aw  = (const float*)d_in[5];  // [C_IN, W_DIM]
  const float* ab  = (const float*)d_in[6];  // [C_IN]
  const float* wt  = (const float*)d_in[7];  // [C_OUT, C_IN, 1, K2]
  const float* bias = (const float*)d_in[8]; // [C_OUT]
  float* out = (float*)d_out;
  float* ws = (float*)d_ws;

  faceconv_prep_kernel<<<dim3(BB), 256, 0, stream>>>(w, aw, ab, wt, ws);
  faceconv_wmma_kernel<<<dim3(GRID_X, BB), 128, 0, stream>>>(x, nbh, ws, bias, out);
}